// SpatialInteractionMix_49452253446339
// MI455X (gfx1250) — compile-verified
//
#include <hip/hip_runtime.h>
#include <math.h>

// Problem constants (fixed by the harness' setup_inputs)
#define BB   16
#define HH   64
#define WW   64
#define CC   192
#define TT   (HH*WW)          // 4096
#define GG   48               // gamma group = C/4
#define NCH  32               // scan chunks per sequence
#define LCH  (TT/NCH)         // 128 tokens per chunk
#define LDSP 194              // padded LDS row stride (even -> 8B aligned b64)
#define TPB1 32               // tokens per workgroup in GEMM kernels

typedef __attribute__((ext_vector_type(2))) float v2f;
typedef __attribute__((ext_vector_type(8))) float v8f;

// ---------------------------------------------------------------------------
// Kernel 0: transpose+prepack Wk,Wv,Wr,Wo into WMMA B-fragment order.
// GEMM is out[m,d] = sum_c A[m,c] * W[d,c]  => B(k=c, n=d) = W[d*C+c].
// Packed layout: packed[(((n*48 + q)*32 + lane)*2 + j)] with
//   c = q*4 + (lane>>4)*2 + j ,  d = n*16 + (lane&15)
// so in the GEMM each lane does ONE contiguous b64 load per K-step.
// ---------------------------------------------------------------------------
__global__ void vrwkv_prepack(const float* __restrict__ Wk,
                              const float* __restrict__ Wv,
                              const float* __restrict__ Wr,
                              const float* __restrict__ Wo,
                              float* __restrict__ packed) {
  int i = blockIdx.x * 256 + threadIdx.x;
  if (i >= 4 * CC * CC) return;
  int m = i / (CC * CC);
  int r = i % (CC * CC);
  int j    = r & 1;
  int lane = (r >> 1) & 31;
  int qn   = r >> 6;
  int q    = qn % 48;      // K chunk (48 * 4 = 192)
  int n    = qn / 48;      // N tile (12 * 16 = 192)
  int c = q * 4 + (lane >> 4) * 2 + j;
  int d = n * 16 + (lane & 15);
  const float* W = (m == 0) ? Wk : (m == 1) ? Wv : (m == 2) ? Wr : Wo;
  packed[i] = W[d * CC + c];
}

// ---------------------------------------------------------------------------
// Kernel 1: q_shift + token mixes -> LDS (32 tokens), then 3 GEMMs (k, v, r)
// with V_WMMA_F32_16X16X4_F32. Each B fragment is reused for 2 token sub-
// tiles (2 independent accumulation chains), halving weight L2 traffic and
// pipelining the matrix unit. sigmoid applied to r on the way out.
// Dynamic LDS: 3 * 32 * LDSP floats = 74,496 B.
// ---------------------------------------------------------------------------
__launch_bounds__(128)
__global__ void vrwkv_mix_gemm(const float* __restrict__ x,
                               const float* __restrict__ mixk,
                               const float* __restrict__ mixv,
                               const float* __restrict__ mixr,
                               const float* __restrict__ packW,   // k,v,r packed
                               float* __restrict__ kbuf,
                               float* __restrict__ vbuf,
                               float* __restrict__ srbuf) {
  extern __shared__ float smem[];
  float (*sxk)[LDSP] = (float(*)[LDSP])(smem);
  float (*sxv)[LDSP] = (float(*)[LDSP])(smem + TPB1 * LDSP);
  float (*sxr)[LDSP] = (float(*)[LDSP])(smem + 2 * TPB1 * LDSP);

  const int t0  = blockIdx.x * TPB1;      // flat token over B*T
  const int b   = t0 / TT;
  const int tt0 = t0 % TT;
  const int h   = tt0 / WW;
  const int w0  = tt0 % WW;               // 32 | w0, tile stays within one h

  // --- shift + mix into LDS ---
  for (int idx = threadIdx.x; idx < TPB1 * CC; idx += 128) {
    int rrow = idx / CC, c = idx % CC;
    int wq = w0 + rrow;
    float xs0 = x[((size_t)b * TT + h * WW + wq) * CC + c];
    float xs  = 0.f;
    if (c < GG)            { if (wq >= 1)      xs = x[((size_t)b*TT + h*WW + wq-1)*CC + c]; }
    else if (c < 2 * GG)   { if (wq <= WW - 2) xs = x[((size_t)b*TT + h*WW + wq+1)*CC + c]; }
    else if (c < 3 * GG)   { if (h  >= 1)      xs = x[((size_t)b*TT + (h-1)*WW + wq)*CC + c]; }
    else                   { if (h  <= HH - 2) xs = x[((size_t)b*TT + (h+1)*WW + wq)*CC + c]; }
    float mk = mixk[c], mv = mixv[c], mr = mixr[c];
    sxk[rrow][c] = xs0 * mk + xs * (1.f - mk);
    sxv[rrow][c] = xs0 * mv + xs * (1.f - mv);
    sxr[rrow][c] = xs0 * mr + xs * (1.f - mr);
  }
  __syncthreads();

  const int wave = threadIdx.x >> 5;
  const int lane = threadIdx.x & 31;
  const int half = lane >> 4;
  const int mrow = lane & 15;

  for (int tix = wave * 9; tix < wave * 9 + 9; ++tix) {
    const int mat = tix / 12;             // 0=k 1=v 2=r
    const int n   = tix % 12;
    const float (*A)[LDSP] = (mat == 0) ? sxk : (mat == 1) ? sxv : sxr;
    const float* pw = packW + (size_t)mat * CC * CC + (size_t)n * 48 * 64;

    v8f acc0 = {};
    v8f acc1 = {};
    for (int q = 0; q < 48; ++q) {
      const int c0 = q * 4 + half * 2;
      v2f bf = *(const v2f*)(pw + q * 64 + lane * 2);          // global b64, coalesced
      v2f a0; a0.x = A[mrow][c0];      a0.y = A[mrow][c0 + 1]; // ds_load_b64
      v2f a1; a1.x = A[16 + mrow][c0]; a1.y = A[16 + mrow][c0 + 1];
      acc0 = __builtin_amdgcn_wmma_f32_16x16x4_f32(
          false, a0, false, bf, (short)0, acc0, false, false);
      acc1 = __builtin_amdgcn_wmma_f32_16x16x4_f32(
          false, a1, false, bf, (short)0, acc1, false, false);
    }
    float* dst = (mat == 0) ? kbuf : (mat == 1) ? vbuf : srbuf;
    for (int tt = 0; tt < 2; ++tt) {
      for (int rl = 0; rl < 8; ++rl) {
        int row = tt * 16 + rl + half * 8;
        float val = (tt == 0) ? acc0[rl] : acc1[rl];
        if (mat == 2) val = __builtin_amdgcn_rcpf(1.f + __expf(-val)); // sigmoid(r)
        dst[((size_t)b * TT + tt0 + row) * CC + n * 16 + mrow] = val;
      }
    }
  }
}

// ---------------------------------------------------------------------------
// Kernel 2a: per-chunk scan summaries (sa, sb, pm) from zero state.
// thread = (b, c, chunk); lanes have consecutive c -> coalesced loads.
// ---------------------------------------------------------------------------
__global__ void vrwkv_wkv_summary(const float* __restrict__ kbuf,
                                  const float* __restrict__ vbuf,
                                  const float* __restrict__ decay,
                                  float* __restrict__ summ) {
  int gid = blockIdx.x * 256 + threadIdx.x;
  if (gid >= BB * CC * NCH) return;
  int c   = gid % CC;
  int rem = gid / CC;
  int b   = rem % BB;
  int ch  = rem / BB;
  float w = decay[c] * (1.f / TT);
  float sa = 0.f, sb = 0.f, pm = -1e38f;
  const float* kp = kbuf + ((size_t)b * TT + ch * LCH) * CC + c;
  const float* vp = vbuf + ((size_t)b * TT + ch * LCH) * CC + c;
  for (int t = 0; t < LCH; ++t) {
    float kt = kp[(size_t)t * CC];
    float vt = vp[(size_t)t * CC];
    float ww2 = pm + w;
    float qq  = fmaxf(ww2, kt);
    float e1  = __expf(ww2 - qq);
    float e2  = __expf(kt - qq);
    sa = e1 * sa + e2 * vt;
    sb = e1 * sb + e2;
    pm = qq;
  }
  size_t si = ((size_t)(b * CC + c) * NCH + ch) * 3;
  summ[si] = sa; summ[si + 1] = sb; summ[si + 2] = pm;
}

// ---------------------------------------------------------------------------
// Kernel 2b: exclusive prefix of chunk states per (b,c):
//   state_{i+1} = decay(state_i, w*L)  (+)  summary_i     (log-max combine)
// ---------------------------------------------------------------------------
__global__ void vrwkv_wkv_prefix(const float* __restrict__ summ,
                                 const float* __restrict__ decay,
                                 float* __restrict__ incoming) {
  int gid = blockIdx.x * 256 + threadIdx.x;
  if (gid >= BB * CC) return;
  int c = gid % CC;
  int b = gid / CC;
  float wL = decay[c] * (1.f / TT) * (float)LCH;
  float aa = 0.f, bb = 0.f, pp = -1e38f;
  for (int ch = 0; ch < NCH; ++ch) {
    size_t idx = ((size_t)(b * CC + c) * NCH + ch) * 3;
    incoming[idx] = aa; incoming[idx + 1] = bb; incoming[idx + 2] = pp;
    float sa = summ[idx], sb = summ[idx + 1], pm = summ[idx + 2];
    float pd = pp + wL;
    float pn = fmaxf(pd, pm);
    float e1 = __expf(pd - pn);
    float e2 = __expf(pm - pn);
    aa = e1 * aa + e2 * sa;
    bb = e1 * bb + e2 * sb;
    pp = pn;
  }
}

// ---------------------------------------------------------------------------
// Kernel 2c: replay each chunk with its incoming state, emit y (exact
// reference step). y overwrites kbuf in place (each thread owns its range).
// Fast v_rcp_f32 shortens the per-step dependent chain.
// ---------------------------------------------------------------------------
__global__ void vrwkv_wkv_apply(float* __restrict__ kbuf,
                                const float* __restrict__ vbuf,
                                const float* __restrict__ decay,
                                const float* __restrict__ first,
                                const float* __restrict__ incoming) {
  int gid = blockIdx.x * 256 + threadIdx.x;
  if (gid >= BB * CC * NCH) return;
  int c   = gid % CC;
  int rem = gid / CC;
  int b   = rem % BB;
  int ch  = rem / BB;
  float w = decay[c] * (1.f / TT);
  float u = first[c] * (1.f / TT);
  size_t idx = ((size_t)(b * CC + c) * NCH + ch) * 3;
  float aa = incoming[idx], bbv = incoming[idx + 1], pp = incoming[idx + 2];
  float*       kp = kbuf + ((size_t)b * TT + ch * LCH) * CC + c;
  const float* vp = vbuf + ((size_t)b * TT + ch * LCH) * CC + c;
  for (int t = 0; t < LCH; ++t) {
    float kt = kp[(size_t)t * CC];
    float vt = vp[(size_t)t * CC];
    float ww = u + kt;
    float p  = fmaxf(pp, ww);
    float e1 = __expf(pp - p);
    float e2 = __expf(ww - p);
    float y  = (e1 * aa + e2 * vt) * __builtin_amdgcn_rcpf(e1 * bbv + e2);
    kp[(size_t)t * CC] = y;
    float ww2 = pp + w;
    float qq  = fmaxf(ww2, kt);
    e1 = __expf(ww2 - qq);
    e2 = __expf(kt - qq);
    aa  = e1 * aa  + e2 * vt;
    bbv = e1 * bbv + e2;
    pp  = qq;
  }
}

// ---------------------------------------------------------------------------
// Kernel 3: per-token LayerNorm over C, gate by sr, then GEMM with Wo (WMMA).
// 32 tokens per workgroup; each B fragment reused by 2 sub-tiles.
// ---------------------------------------------------------------------------
__launch_bounds__(128)
__global__ void vrwkv_ln_out(const float* __restrict__ ybuf,
                             const float* __restrict__ srbuf,
                             const float* __restrict__ lng,
                             const float* __restrict__ lnb,
                             const float* __restrict__ packWo,
                             float* __restrict__ out) {
  __shared__ float sa[TPB1][LDSP];
  __shared__ float smu[TPB1];
  __shared__ float srs[TPB1];

  const size_t t0 = (size_t)blockIdx.x * TPB1;   // flat over B*T

  for (int idx = threadIdx.x; idx < TPB1 * CC; idx += 128) {
    int r = idx / CC, c = idx % CC;
    sa[r][c] = ybuf[(t0 + r) * CC + c];
  }
  __syncthreads();

  if (threadIdx.x < TPB1) {
    int r = threadIdx.x;
    float s = 0.f;
    for (int c = 0; c < CC; ++c) s += sa[r][c];
    float mu = s * (1.f / CC);
    float v = 0.f;
    for (int c = 0; c < CC; ++c) { float d = sa[r][c] - mu; v += d * d; }
    smu[r] = mu;
    srs[r] = __builtin_amdgcn_rsqf(v * (1.f / CC) + 1e-5f);
  }
  __syncthreads();

  for (int idx = threadIdx.x; idx < TPB1 * CC; idx += 128) {
    int r = idx / CC, c = idx % CC;
    float nv = (sa[r][c] - smu[r]) * srs[r] * lng[c] + lnb[c];
    sa[r][c] = nv * srbuf[(t0 + r) * CC + c];
  }
  __syncthreads();

  const int wave = threadIdx.x >> 5;
  const int lane = threadIdx.x & 31;
  const int half = lane >> 4;
  const int mrow = lane & 15;

  for (int n = wave * 3; n < wave * 3 + 3; ++n) {
    const float* pw = packWo + (size_t)n * 48 * 64;
    v8f acc0 = {};
    v8f acc1 = {};
    for (int q = 0; q < 48; ++q) {
      const int c0 = q * 4 + half * 2;
      v2f bf = *(const v2f*)(pw + q * 64 + lane * 2);
      v2f a0; a0.x = sa[mrow][c0];      a0.y = sa[mrow][c0 + 1];
      v2f a1; a1.x = sa[16 + mrow][c0]; a1.y = sa[16 + mrow][c0 + 1];
      acc0 = __builtin_amdgcn_wmma_f32_16x16x4_f32(
          false, a0, false, bf, (short)0, acc0, false, false);
      acc1 = __builtin_amdgcn_wmma_f32_16x16x4_f32(
          false, a1, false, bf, (short)0, acc1, false, false);
    }
    for (int tt = 0; tt < 2; ++tt) {
      for (int rl = 0; rl < 8; ++rl) {
        int row = tt * 16 + rl + half * 8;
        float val = (tt == 0) ? acc0[rl] : acc1[rl];
        out[(t0 + row) * CC + n * 16 + mrow] = val;
      }
    }
  }
}

// ---------------------------------------------------------------------------
extern "C" void kernel_launch(void* const* d_in, const int* in_sizes, int n_in,
                              void* d_out, int out_size, void* d_ws, size_t ws_size,
                              hipStream_t stream) {
  const float* x     = (const float*)d_in[0];
  const float* decay = (const float*)d_in[3];
  const float* first = (const float*)d_in[4];
  const float* mixk  = (const float*)d_in[5];
  const float* mixv  = (const float*)d_in[6];
  const float* mixr  = (const float*)d_in[7];
  const float* Wk    = (const float*)d_in[8];
  const float* Wv    = (const float*)d_in[9];
  const float* Wr    = (const float*)d_in[10];
  const float* Wo    = (const float*)d_in[11];
  const float* lng   = (const float*)d_in[12];
  const float* lnb   = (const float*)d_in[13];

  const size_t BTC = (size_t)BB * TT * CC;
  float* packW    = (float*)d_ws;                 // 4*C*C
  float* kbuf     = packW + 4 * CC * CC;          // BTC (later holds y)
  float* vbuf     = kbuf + BTC;                   // BTC
  float* srbuf    = vbuf + BTC;                   // BTC
  float* summ     = srbuf + BTC;                  // B*C*NCH*3
  float* incoming = summ + (size_t)BB * CC * NCH * 3;

  vrwkv_prepack<<<(4 * CC * CC + 255) / 256, 256, 0, stream>>>(Wk, Wv, Wr, Wo, packW);

  vrwkv_mix_gemm<<<(BB * TT) / TPB1, 128, 3 * TPB1 * LDSP * sizeof(float), stream>>>(
      x, mixk, mixv, mixr, packW, kbuf, vbuf, srbuf);

  vrwkv_wkv_summary<<<(BB * CC * NCH + 255) / 256, 256, 0, stream>>>(
      kbuf, vbuf, decay, summ);

  vrwkv_wkv_prefix<<<(BB * CC + 255) / 256, 256, 0, stream>>>(
      summ, decay, incoming);

  vrwkv_wkv_apply<<<(BB * CC * NCH + 255) / 256, 256, 0, stream>>>(
      kbuf, vbuf, decay, first, incoming);

  vrwkv_ln_out<<<(BB * TT) / TPB1, 128, 0, stream>>>(
      kbuf, srbuf, lng, lnb, packW + 3 * (size_t)CC * CC, (float*)d_out);
}